// LinearSSMLayer_47854525612488
// MI455X (gfx1250) — compile-verified
//
#include <hip/hip_runtime.h>
#include <hip/hip_bf16.h>

#define D_MODEL 1024
#define D_STATE 16
#define BATCH   2
#define SEQ     2048
#define NCHUNK  16
#define LCHUNK  (SEQ / NCHUNK)   // 128

typedef __attribute__((ext_vector_type(2))) float v2f;
typedef __attribute__((ext_vector_type(8))) float v8f;

// ---------------------------------------------------------------------------
// K0: A_bar[d,s] = exp(-exp(logA[d,s]));  A_Lc = A_bar^LCHUNK (7 squarings)
// ---------------------------------------------------------------------------
__global__ void ssm_abar_kernel(const float* __restrict__ logA,
                                float* __restrict__ A_bar,
                                float* __restrict__ A_Lc) {
  int i = blockIdx.x * blockDim.x + threadIdx.x;
  if (i < D_MODEL * D_STATE) {
    float ab = expf(-expf(logA[i]));
    A_bar[i] = ab;
    float p = ab;
#pragma unroll
    for (int j = 0; j < 7; ++j) p = p * p;   // ab^128
    A_Lc[i] = p;
  }
}

// ---------------------------------------------------------------------------
// K1: Bt[row, s] = sum_k x[row, k] * B_w[s, k]   (row = b*SEQ + l)
// fp32 WMMA 16x16x4: one wave computes a 16(rows) x 16(states) tile.
// A frag (16x4 f32): lane m=lane&15, khalf=lane>>4 holds {K=2*khalf, 2*khalf+1}
// B frag (4x16 f32): lane n=lane&15, khalf holds B[k][n] = B_w[n][k] pairs.
// Both fragments are contiguous float2 loads from row-major memory.
// ---------------------------------------------------------------------------
__global__ void ssm_bproj_wmma_kernel(const float* __restrict__ x,
                                      const float* __restrict__ Bw,
                                      float* __restrict__ Bt) {
  const int row0 = blockIdx.x * 16;
  const int lane = threadIdx.x;
  const int half = lane >> 4;     // 0 or 1
  const int lidx = lane & 15;

  const float* arow = x  + (size_t)(row0 + lidx) * D_MODEL;
  const float* brow = Bw + (size_t)lidx * D_MODEL;

  v8f acc = {};
  for (int k0 = 0; k0 < D_MODEL; k0 += 4) {
    v2f a, b;
    a.x = arow[k0 + 2 * half + 0];
    a.y = arow[k0 + 2 * half + 1];
    b.x = brow[k0 + 2 * half + 0];
    b.y = brow[k0 + 2 * half + 1];
    // 8 args: (neg_a, A, neg_b, B, c_mod, C, reuse_a, reuse_b)
    acc = __builtin_amdgcn_wmma_f32_16x16x4_f32(false, a, false, b,
                                                (short)0, acc, false, false);
  }
  // D layout: VGPR i holds M = i + 8*half, N = lidx
#pragma unroll
  for (int i = 0; i < 8; ++i) {
    Bt[(size_t)(row0 + i + 8 * half) * D_STATE + lidx] = acc[i];
  }
}

// ---------------------------------------------------------------------------
// K2: per-chunk local scan from h=0; emit chunk-final state.
// thread <-> (b, d, chunk); 16 independent state chains in registers.
// ---------------------------------------------------------------------------
__global__ void ssm_chunk_local_kernel(const float* __restrict__ Bt,
                                       const float* __restrict__ A_bar,
                                       float* __restrict__ Ssum) {
  const int d = blockIdx.x * blockDim.x + threadIdx.x;
  const int c = blockIdx.y;
  const int b = blockIdx.z;

  float A[D_STATE], h[D_STATE];
  {
    const float4* p = reinterpret_cast<const float4*>(A_bar + (size_t)d * D_STATE);
#pragma unroll
    for (int q = 0; q < 4; ++q) {
      float4 v = p[q];
      A[4 * q + 0] = v.x; A[4 * q + 1] = v.y;
      A[4 * q + 2] = v.z; A[4 * q + 3] = v.w;
    }
  }
#pragma unroll
  for (int s = 0; s < D_STATE; ++s) h[s] = 0.0f;

  const float* btrow = Bt + ((size_t)b * SEQ + (size_t)c * LCHUNK) * D_STATE;
  for (int l = 0; l < LCHUNK; ++l) {
    const float4* p = reinterpret_cast<const float4*>(btrow + (size_t)l * D_STATE);
#pragma unroll
    for (int q = 0; q < 4; ++q) {
      float4 v = p[q];
      h[4 * q + 0] = fmaf(A[4 * q + 0], h[4 * q + 0], v.x);
      h[4 * q + 1] = fmaf(A[4 * q + 1], h[4 * q + 1], v.y);
      h[4 * q + 2] = fmaf(A[4 * q + 2], h[4 * q + 2], v.z);
      h[4 * q + 3] = fmaf(A[4 * q + 3], h[4 * q + 3], v.w);
    }
  }

  float4* out = reinterpret_cast<float4*>(
      Ssum + (((size_t)b * D_MODEL + d) * NCHUNK + c) * D_STATE);
#pragma unroll
  for (int q = 0; q < 4; ++q) {
    float4 v = { h[4 * q + 0], h[4 * q + 1], h[4 * q + 2], h[4 * q + 3] };
    out[q] = v;
  }
}

// ---------------------------------------------------------------------------
// K3: sequential prefix over the 16 chunk carries (tiny).
// Hinit[b,d,c,s] = state entering chunk c.
// ---------------------------------------------------------------------------
__global__ void ssm_prefix_kernel(const float* __restrict__ Ssum,
                                  const float* __restrict__ A_Lc,
                                  float* __restrict__ Hinit) {
  int idx = blockIdx.x * blockDim.x + threadIdx.x;   // b*D + d
  if (idx >= BATCH * D_MODEL) return;
  int d = idx & (D_MODEL - 1);

  float AL[D_STATE], h[D_STATE];
  {
    const float4* p = reinterpret_cast<const float4*>(A_Lc + (size_t)d * D_STATE);
#pragma unroll
    for (int q = 0; q < 4; ++q) {
      float4 v = p[q];
      AL[4 * q + 0] = v.x; AL[4 * q + 1] = v.y;
      AL[4 * q + 2] = v.z; AL[4 * q + 3] = v.w;
    }
  }
#pragma unroll
  for (int s = 0; s < D_STATE; ++s) h[s] = 0.0f;

  for (int c = 0; c < NCHUNK; ++c) {
    size_t off = ((size_t)idx * NCHUNK + c) * D_STATE;
    float4* ho = reinterpret_cast<float4*>(Hinit + off);
    const float4* si = reinterpret_cast<const float4*>(Ssum + off);
#pragma unroll
    for (int q = 0; q < 4; ++q) {
      float4 v = { h[4 * q + 0], h[4 * q + 1], h[4 * q + 2], h[4 * q + 3] };
      ho[q] = v;
      float4 s4 = si[q];
      h[4 * q + 0] = fmaf(AL[4 * q + 0], h[4 * q + 0], s4.x);
      h[4 * q + 1] = fmaf(AL[4 * q + 1], h[4 * q + 1], s4.y);
      h[4 * q + 2] = fmaf(AL[4 * q + 2], h[4 * q + 2], s4.z);
      h[4 * q + 3] = fmaf(AL[4 * q + 3], h[4 * q + 3], s4.w);
    }
  }
}

// ---------------------------------------------------------------------------
// K4: replay each chunk from its correct initial state; fuse C-projection and
// skip connection. Lanes <-> consecutive d => coalesced x/y, broadcast Bt row.
// ---------------------------------------------------------------------------
__global__ void ssm_output_kernel(const float* __restrict__ x,
                                  const float* __restrict__ Bt,
                                  const float* __restrict__ A_bar,
                                  const float* __restrict__ Cw,
                                  const float* __restrict__ Dvec,
                                  const float* __restrict__ Hinit,
                                  float* __restrict__ y) {
  const int d = blockIdx.x * blockDim.x + threadIdx.x;
  const int c = blockIdx.y;
  const int b = blockIdx.z;

  float A[D_STATE], C[D_STATE], h[D_STATE];
  {
    const float4* pa = reinterpret_cast<const float4*>(A_bar + (size_t)d * D_STATE);
    const float4* pc = reinterpret_cast<const float4*>(Cw + (size_t)d * D_STATE);
    const float4* ph = reinterpret_cast<const float4*>(
        Hinit + (((size_t)b * D_MODEL + d) * NCHUNK + c) * D_STATE);
#pragma unroll
    for (int q = 0; q < 4; ++q) {
      float4 va = pa[q], vc = pc[q], vh = ph[q];
      A[4 * q + 0] = va.x; A[4 * q + 1] = va.y; A[4 * q + 2] = va.z; A[4 * q + 3] = va.w;
      C[4 * q + 0] = vc.x; C[4 * q + 1] = vc.y; C[4 * q + 2] = vc.z; C[4 * q + 3] = vc.w;
      h[4 * q + 0] = vh.x; h[4 * q + 1] = vh.y; h[4 * q + 2] = vh.z; h[4 * q + 3] = vh.w;
    }
  }
  const float dskip = Dvec[d];

  const size_t rowbase = (size_t)b * SEQ + (size_t)c * LCHUNK;
  const float* btrow = Bt + rowbase * D_STATE;
  const float* xcol  = x + rowbase * D_MODEL + d;
  float*       ycol  = y + rowbase * D_MODEL + d;

  for (int l = 0; l < LCHUNK; ++l) {
    if (l + 16 < LCHUNK)
      __builtin_prefetch(xcol + (size_t)(l + 16) * D_MODEL, 0, 0);

    const float4* p = reinterpret_cast<const float4*>(btrow + (size_t)l * D_STATE);
    float acc = 0.0f;
#pragma unroll
    for (int q = 0; q < 4; ++q) {
      float4 v = p[q];
      h[4 * q + 0] = fmaf(A[4 * q + 0], h[4 * q + 0], v.x);
      h[4 * q + 1] = fmaf(A[4 * q + 1], h[4 * q + 1], v.y);
      h[4 * q + 2] = fmaf(A[4 * q + 2], h[4 * q + 2], v.z);
      h[4 * q + 3] = fmaf(A[4 * q + 3], h[4 * q + 3], v.w);
      acc = fmaf(C[4 * q + 0], h[4 * q + 0], acc);
      acc = fmaf(C[4 * q + 1], h[4 * q + 1], acc);
      acc = fmaf(C[4 * q + 2], h[4 * q + 2], acc);
      acc = fmaf(C[4 * q + 3], h[4 * q + 3], acc);
    }
    float xv = xcol[(size_t)l * D_MODEL];
    ycol[(size_t)l * D_MODEL] = fmaf(dskip, xv, acc);
  }
}

// ---------------------------------------------------------------------------
extern "C" void kernel_launch(void* const* d_in, const int* in_sizes, int n_in,
                              void* d_out, int out_size, void* d_ws, size_t ws_size,
                              hipStream_t stream) {
  const float* x    = (const float*)d_in[0];   // (B, L, D)
  const float* logA = (const float*)d_in[1];   // (D, S)
  const float* Bw   = (const float*)d_in[2];   // (S, D)
  const float* Cw   = (const float*)d_in[3];   // (D, S)
  const float* Dv   = (const float*)d_in[4];   // (D,)
  float* y = (float*)d_out;                    // (B, L, D)

  char* ws = (char*)d_ws;
  float* A_bar = (float*)(ws);                              // 64 KB
  float* A_Lc  = (float*)(ws + (64 << 10));                 // 64 KB
  float* Bt    = (float*)(ws + (128 << 10));                // 256 KB
  float* Ssum  = (float*)(ws + (384 << 10));                // 2 MB
  float* Hinit = (float*)(ws + (384 << 10) + (2 << 20));    // 2 MB

  // K0: decay factors
  ssm_abar_kernel<<<(D_MODEL * D_STATE + 255) / 256, 256, 0, stream>>>(
      logA, A_bar, A_Lc);

  // K1: B projection via fp32 WMMA (one wave per 16x16 tile)
  ssm_bproj_wmma_kernel<<<(BATCH * SEQ) / 16, 32, 0, stream>>>(x, Bw, Bt);

  // K2/K4 grid: lanes over d (coalesced), chunks, batch
  dim3 gscan(D_MODEL / 256, NCHUNK, BATCH);

  // K2: chunk-local scans
  ssm_chunk_local_kernel<<<gscan, 256, 0, stream>>>(Bt, A_bar, Ssum);

  // K3: carry prefix across chunks
  ssm_prefix_kernel<<<(BATCH * D_MODEL + 255) / 256, 256, 0, stream>>>(
      Ssum, A_Lc, Hinit);

  // K4: replay + C-projection + skip
  ssm_output_kernel<<<gscan, 256, 0, stream>>>(x, Bt, A_bar, Cw, Dv, Hinit, y);
}